// LatentLongRange_27943057228071
// MI455X (gfx1250) — compile-verified
//
#include <hip/hip_runtime.h>
#include <math.h>

#define N_ATOMS 6144
#define F_DIM   128
#define NB_SEG  48
#define NCH     (N_ATOMS / 16)          // 384 row/col chunks
#define TILES   (NCH * NCH)             // 147456 16x16 tiles
#define WPB     8                       // waves per block (wave32)
#define PAIR_BLOCKS (TILES / WPB)       // 18432

typedef __attribute__((ext_vector_type(2))) float v2f;
typedef __attribute__((ext_vector_type(8))) float v8f;

// ---------------------------------------------------------------------------
// Stage 1: per-atom features. One wave (32 lanes) per atom.
//   q_raw = silu(h0@qW1+qb1)@qW2 + qb2
//   c6s   = sqrt(softplus(silu(h0@cW1+cb1)@cW2 + cb2))
//   mu    = h1 @ muW            (3-vector)
//   pp    = |pos|^2,  mup = mu . pos
// ---------------------------------------------------------------------------
__global__ void __launch_bounds__(256) atom_feats(
    const float* __restrict__ h0, const float* __restrict__ h1,
    const float* __restrict__ pos,
    const float* __restrict__ qW1, const float* __restrict__ qb1,
    const float* __restrict__ qW2, const float* __restrict__ qb2,
    const float* __restrict__ cW1, const float* __restrict__ cb1,
    const float* __restrict__ cW2, const float* __restrict__ cb2,
    const float* __restrict__ muW,
    float* __restrict__ qraw, float* __restrict__ c6s,
    float* __restrict__ mu, float* __restrict__ pp, float* __restrict__ mup)
{
    const int atom = (blockIdx.x * blockDim.x + threadIdx.x) >> 5;
    const int lane = threadIdx.x & 31;
    if (atom >= N_ATOMS) return;

    const float* hrow = h0 + atom * F_DIM;

    // hidden layer: 64 cols, 2 per lane (j = lane, lane+32)
    float aq0 = qb1[lane], aq1 = qb1[lane + 32];
    float ac0 = cb1[lane], ac1 = cb1[lane + 32];
    for (int k = 0; k < F_DIM; ++k) {
        const float x = hrow[k];
        const float* wq = qW1 + k * 64;
        const float* wc = cW1 + k * 64;
        aq0 += x * wq[lane]; aq1 += x * wq[lane + 32];
        ac0 += x * wc[lane]; ac1 += x * wc[lane + 32];
    }
    // silu (hardware rcp: ~1ulp, fine for features feeding an energy sum)
    const float sq0 = aq0 * __builtin_amdgcn_rcpf(1.f + __expf(-aq0));
    const float sq1 = aq1 * __builtin_amdgcn_rcpf(1.f + __expf(-aq1));
    const float sc0 = ac0 * __builtin_amdgcn_rcpf(1.f + __expf(-ac0));
    const float sc1 = ac1 * __builtin_amdgcn_rcpf(1.f + __expf(-ac1));
    float pq = sq0 * qW2[lane] + sq1 * qW2[lane + 32];
    float pc = sc0 * cW2[lane] + sc1 * cW2[lane + 32];

    // mu partials across k
    float m0 = 0.f, m1 = 0.f, m2 = 0.f;
    const float* h1r = h1 + atom * 3 * F_DIM;
    for (int k = lane; k < F_DIM; k += 32) {
        const float w = muW[k];
        m0 += h1r[k]         * w;
        m1 += h1r[128 + k]   * w;
        m2 += h1r[256 + k]   * w;
    }

    // wave32 reductions
    for (int off = 16; off > 0; off >>= 1) {
        pq += __shfl_down(pq, off, 32);
        pc += __shfl_down(pc, off, 32);
        m0 += __shfl_down(m0, off, 32);
        m1 += __shfl_down(m1, off, 32);
        m2 += __shfl_down(m2, off, 32);
    }

    if (lane == 0) {
        const float qv = pq + qb2[0];
        const float cv = pc + cb2[0];
        // numerically stable softplus
        const float sp = fmaxf(cv, 0.f) + log1pf(__expf(-fabsf(cv)));
        qraw[atom] = qv;
        c6s[atom]  = __builtin_amdgcn_sqrtf(sp);   // c6_ij = sqrt(c6_i)*sqrt(c6_j)
        mu[atom * 3 + 0] = m0;
        mu[atom * 3 + 1] = m1;
        mu[atom * 3 + 2] = m2;
        const float* p = pos + atom * 3;
        pp[atom]  = p[0]*p[0] + p[1]*p[1] + p[2]*p[2];
        mup[atom] = m0*p[0] + m1*p[1] + m2*p[2];
    }
}

// ---------------------------------------------------------------------------
// Stage 2a: per-molecule mean charge (scatter_mean). One block per segment.
// ---------------------------------------------------------------------------
__global__ void __launch_bounds__(256) seg_mean(
    const float* __restrict__ qraw, const int* __restrict__ batch,
    float* __restrict__ qmean)
{
    const int b = blockIdx.x;
    float s = 0.f, c = 0.f;
    for (int i = threadIdx.x; i < N_ATOMS; i += 256) {
        if (batch[i] == b) { s += qraw[i]; c += 1.f; }
    }
    __shared__ float ss[256], cc[256];
    ss[threadIdx.x] = s; cc[threadIdx.x] = c;
    __syncthreads();
    for (int off = 128; off > 0; off >>= 1) {
        if (threadIdx.x < off) {
            ss[threadIdx.x] += ss[threadIdx.x + off];
            cc[threadIdx.x] += cc[threadIdx.x + off];
        }
        __syncthreads();
    }
    if (threadIdx.x == 0) qmean[b] = ss[0] / fmaxf(cc[0], 1.f);
}

// Stage 2b: enforce per-molecule neutrality in place.
__global__ void __launch_bounds__(256) neutralize(
    float* __restrict__ qraw, const int* __restrict__ batch,
    const float* __restrict__ qmean)
{
    const int i = blockIdx.x * blockDim.x + threadIdx.x;
    if (i < N_ATOMS) qraw[i] -= qmean[batch[i]];
}

// ---------------------------------------------------------------------------
// Stage 3: pairwise energy over 16x16 tiles, one tile per wave.
// Gram matrices via V_WMMA_F32_16X16X4_F32 (K=3 padded to 4, full f32).
// Tiles with non-overlapping sorted-batch ranges are skipped (~98%).
// Inner loop is branchless (validf multiply) with hardware sqrt/rcp.
// ---------------------------------------------------------------------------
__global__ void __launch_bounds__(256) pair_energy(
    const float* __restrict__ pos, const int* __restrict__ batch,
    const float* __restrict__ qc, const float* __restrict__ c6s,
    const float* __restrict__ mu, const float* __restrict__ pp,
    const float* __restrict__ mup,
    float* __restrict__ partial)
{
    const int w    = threadIdx.x >> 5;      // wave in block
    const int lane = threadIdx.x & 31;
    const int tile = blockIdx.x * WPB + w;
    const int ti = tile / NCH, tj = tile % NCH;
    const int i0 = ti * 16,  j0 = tj * 16;

    __shared__ float sRow[WPB][4][16];      // pp, q, c6s, mup (rows i)
    __shared__ float sCol[WPB][4][16];      // pp, q, c6s, mup (cols j)
    __shared__ int   sRowB[WPB][16];
    __shared__ int   sColB[WPB][16];
    __shared__ float blkSum[WPB];

    float esum = 0.f;

    // wave-uniform early-out: batch is sorted, tile valid only if ranges overlap
    const bool active = !(batch[i0] > batch[j0 + 15] || batch[j0] > batch[i0 + 15]);

    if (active) {
        // stage per-tile scalars into this wave's LDS slab
        if (lane < 16) {
            const int gi = i0 + lane;
            sRow[w][0][lane] = pp[gi];
            sRow[w][1][lane] = qc[gi];
            sRow[w][2][lane] = c6s[gi];
            sRow[w][3][lane] = mup[gi];
            sRowB[w][lane]   = batch[gi];
        } else {
            const int gj = j0 + (lane - 16);
            sCol[w][0][lane - 16] = pp[gj];
            sCol[w][1][lane - 16] = qc[gj];
            sCol[w][2][lane - 16] = c6s[gj];
            sCol[w][3][lane - 16] = mup[gj];
            sColB[w][lane - 16]   = batch[gj];
        }

        // Build WMMA operands. 16x4 f32 A layout (wave32):
        //   lanes 0-15:  VGPR0=K0, VGPR1=K1 ; lanes 16-31: VGPR0=K2, VGPR1=K3
        // Load all 3 components unconditionally -> wide load + v_cndmask,
        // no exec-mask branches around the loads.
        const int lm = lane & 15;
        const int hi = lane >> 4;
        const float* pi = pos + (i0 + lm) * 3;
        const float* pj = pos + (j0 + lm) * 3;
        const float* mi = mu  + (i0 + lm) * 3;
        const float* mj = mu  + (j0 + lm) * 3;
        const float pix = pi[0], piy = pi[1], piz = pi[2];
        const float pjx = pj[0], pjy = pj[1], pjz = pj[2];
        const float mix = mi[0], miy = mi[1], miz = mi[2];
        const float mjx = mj[0], mjy = mj[1], mjz = mj[2];

        v2f a_pos, b_pos, a_mu, b_mu;
        a_pos.x = hi ? piz : pix;  a_pos.y = hi ? 0.f : piy;
        b_pos.x = hi ? pjz : pjx;  b_pos.y = hi ? 0.f : pjy;
        a_mu.x  = hi ? miz : mix;  a_mu.y  = hi ? 0.f : miy;
        b_mu.x  = hi ? mjz : mjx;  b_mu.y  = hi ? 0.f : mjy;

        v8f z = {0.f, 0.f, 0.f, 0.f, 0.f, 0.f, 0.f, 0.f};
        // D = A x B (+0): four 16x16 Gram matrices, full f32
        v8f Gpp = __builtin_amdgcn_wmma_f32_16x16x4_f32(false, a_pos, false, b_pos, (short)0, z, false, false);
        v8f Gmm = __builtin_amdgcn_wmma_f32_16x16x4_f32(false, a_mu,  false, b_mu,  (short)0, z, false, false);
        v8f Gmp = __builtin_amdgcn_wmma_f32_16x16x4_f32(false, a_mu,  false, b_pos, (short)0, z, false, false); // mu_i . p_j
        v8f Gpm = __builtin_amdgcn_wmma_f32_16x16x4_f32(false, a_pos, false, b_mu,  (short)0, z, false, false); // p_i . mu_j

        // this lane owns column n = lm, rows m = v + 8*hi (C/D f32 layout)
        const float cPP = sCol[w][0][lm], cQ = sCol[w][1][lm];
        const float cC  = sCol[w][2][lm], cMP = sCol[w][3][lm];
        const int   cB  = sColB[w][lm];
        const int   gj  = j0 + lm;

        #pragma unroll
        for (int v = 0; v < 8; ++v) {
            const int m  = v + 8 * hi;
            const int gi = i0 + m;
            const float validf = (sRowB[w][m] == cB && gi != gj) ? 1.f : 0.f;

            const float rPP = sRow[w][0][m], rQ = sRow[w][1][m];
            const float rC  = sRow[w][2][m], rMP = sRow[w][3][m];

            // clamp: rank-1 rearrangement can round slightly negative at i==j;
            // keeps branchless path NaN-free (0 * NaN would poison the sum)
            const float d2    = fmaxf(rPP + cPP - 2.f * Gpp[v], 0.f);
            const float dist  = __builtin_amdgcn_sqrtf(d2 + 1e-8f);
            const float inv_d = __builtin_amdgcn_rcpf(dist);

            // Coulomb
            const float taper = 1.f - __expf(-0.5f * dist);
            const float ec = rQ * cQ * taper * inv_d;
            // van der Waals
            const float r6 = d2 * d2 * d2;
            const float ev = -(rC * cC) * __builtin_amdgcn_rcpf(r6 + 20.f);
            // dipole-dipole
            const float invd2 = __builtin_amdgcn_rcpf(dist + 1e-8f);
            const float mdni = (rMP - Gmp[v]) * invd2;
            const float mdnj = (Gpm[v] - cMP) * invd2;
            const float ed = (Gmm[v] - 3.f * mdni * mdnj)
                           * __builtin_amdgcn_rcpf(d2 * dist + 10.f);

            // 0.5*14.399 = 7.1995 folded
            esum += validf * (7.1995f * ec + 0.5f * ev + 0.5f * ed);
        }
    }

    // deterministic reduction: wave -> block -> per-block partial
    for (int off = 16; off > 0; off >>= 1) esum += __shfl_down(esum, off, 32);
    if (lane == 0) blkSum[w] = esum;
    __syncthreads();
    if (threadIdx.x == 0) {
        float s = 0.f;
        for (int k = 0; k < WPB; ++k) s += blkSum[k];
        partial[blockIdx.x] = s;
    }
}

// ---------------------------------------------------------------------------
// Stage 4: deterministic final reduction of per-block partials.
// ---------------------------------------------------------------------------
__global__ void __launch_bounds__(256) final_reduce(
    const float* __restrict__ partial, int n, float* __restrict__ out)
{
    __shared__ float s[256];
    float acc = 0.f;
    for (int i = threadIdx.x; i < n; i += 256) acc += partial[i];
    s[threadIdx.x] = acc;
    __syncthreads();
    for (int off = 128; off > 0; off >>= 1) {
        if (threadIdx.x < off) s[threadIdx.x] += s[threadIdx.x + off];
        __syncthreads();
    }
    if (threadIdx.x == 0) out[0] = s[0] * 1.0f;   // LONG_RANGE_SCALE
}

// ---------------------------------------------------------------------------
extern "C" void kernel_launch(void* const* d_in, const int* in_sizes, int n_in,
                              void* d_out, int out_size, void* d_ws, size_t ws_size,
                              hipStream_t stream) {
    (void)in_sizes; (void)n_in; (void)out_size; (void)ws_size;

    const float* h0   = (const float*)d_in[0];
    const float* h1   = (const float*)d_in[1];
    const float* pos  = (const float*)d_in[2];
    const int*   bat  = (const int*)  d_in[3];
    const float* qW1  = (const float*)d_in[4];
    const float* qb1  = (const float*)d_in[5];
    const float* qW2  = (const float*)d_in[6];
    const float* qb2  = (const float*)d_in[7];
    const float* cW1  = (const float*)d_in[8];
    const float* cb1  = (const float*)d_in[9];
    const float* cW2  = (const float*)d_in[10];
    const float* cb2  = (const float*)d_in[11];
    const float* muW  = (const float*)d_in[12];
    float* out = (float*)d_out;

    // workspace layout (floats)
    float* ws     = (float*)d_ws;
    float* qraw   = ws;                    // N
    float* c6s    = qraw + N_ATOMS;        // N
    float* mu     = c6s  + N_ATOMS;        // 3N
    float* pp     = mu   + 3 * N_ATOMS;    // N
    float* mup    = pp   + N_ATOMS;        // N
    float* qmean  = mup  + N_ATOMS;        // NB
    float* parts  = qmean + NB_SEG;        // PAIR_BLOCKS

    atom_feats<<<N_ATOMS / 8, 256, 0, stream>>>(
        h0, h1, pos, qW1, qb1, qW2, qb2, cW1, cb1, cW2, cb2, muW,
        qraw, c6s, mu, pp, mup);

    seg_mean<<<NB_SEG, 256, 0, stream>>>(qraw, bat, qmean);
    neutralize<<<N_ATOMS / 256, 256, 0, stream>>>(qraw, bat, qmean);

    pair_energy<<<PAIR_BLOCKS, 256, 0, stream>>>(
        pos, bat, qraw, c6s, mu, pp, mup, parts);

    final_reduce<<<1, 256, 0, stream>>>(parts, PAIR_BLOCKS, out);
}